// MultiHeadedAttention_41377714929886
// MI455X (gfx1250) — compile-verified
//
#include <hip/hip_runtime.h>
#include <hip/hip_bf16.h>

#define LQ_  1024
#define LKV_ 1024
#define DD   128
#define BB   2

typedef float v2f __attribute__((ext_vector_type(2)));
typedef float v8f __attribute__((ext_vector_type(8)));

// ---- native tanh, 4-wide ----------------------------------------------------
// Prefer the clang builtin (compiler manages trans-op hazards). If absent,
// emit v_tanh_f32 directly: 4 independent tanh's back-to-back (each is the
// required "1 independent op" after the previous trans op) + trailing v_nop
// to cover the last result before its consumer (ISA 5.7.4 trans hazard).
#if __has_builtin(__builtin_amdgcn_tanhf)
#warning "CDNA5 tanh path: __builtin_amdgcn_tanhf (native v_tanh_f32)"
__device__ __forceinline__ float4 dev_tanh4(float4 x) {
    float4 r;
    r.x = __builtin_amdgcn_tanhf(x.x);
    r.y = __builtin_amdgcn_tanhf(x.y);
    r.z = __builtin_amdgcn_tanhf(x.z);
    r.w = __builtin_amdgcn_tanhf(x.w);
    return r;
}
#elif __has_builtin(__builtin_amdgcn_tanh_f32)
#warning "CDNA5 tanh path: __builtin_amdgcn_tanh_f32 (native v_tanh_f32)"
__device__ __forceinline__ float4 dev_tanh4(float4 x) {
    float4 r;
    r.x = __builtin_amdgcn_tanh_f32(x.x);
    r.y = __builtin_amdgcn_tanh_f32(x.y);
    r.z = __builtin_amdgcn_tanh_f32(x.z);
    r.w = __builtin_amdgcn_tanh_f32(x.w);
    return r;
}
#else
#warning "CDNA5 tanh path: inline-asm v_tanh_f32 (builtin not declared)"
__device__ __forceinline__ float4 dev_tanh4(float4 x) {
    float4 r;
    asm("v_tanh_f32 %0, %4\n\t"
        "v_tanh_f32 %1, %5\n\t"
        "v_tanh_f32 %2, %6\n\t"
        "v_tanh_f32 %3, %7\n\t"
        "v_nop"
        : "=&v"(r.x), "=&v"(r.y), "=&v"(r.z), "=&v"(r.w)
        : "v"(x.x), "v"(x.y), "v"(x.z), "v"(x.w));
    return r;
}
#endif

__device__ __forceinline__ float dev_exp(float x) {
    return __builtin_amdgcn_exp2f(x * 1.44269504088896340736f);   // v_exp_f32
}

// ---------------------------------------------------------------------------
// Kernel 0: fold the projection chains.
//   Aq[f,d] = (sum_e Wq[e,f] * Wc[d,e]) / sqrt(D)   biasq[d] = (bq@WcQ)/sqrt(D) + bc[d]
//   Ak[f,d] =  sum_e Wk[e,f] * Wc[d,D+e]            biask[d] =  bk@WcK
// 2 x 128^3 GEMM = 4 MFLOP total; plain VALU is fine.
// ---------------------------------------------------------------------------
__global__ __launch_bounds__(DD) void combine_weights_kernel(
    const float* __restrict__ Wq, const float* __restrict__ bq,
    const float* __restrict__ Wk, const float* __restrict__ bk,
    const float* __restrict__ Wc, const float* __restrict__ bc,
    float* __restrict__ Aq, float* __restrict__ Ak,
    float* __restrict__ biasq, float* __restrict__ biask) {
    const int sel = blockIdx.y;      // 0 = q-side, 1 = k-side
    const int f   = blockIdx.x;      // input-feature row of A
    const int d   = threadIdx.x;     // output column of A
    const float scale = sel ? 1.0f : 0.08838834764831844f; // 1/sqrt(128)
    const float* W    = sel ? Wk : Wq;
    const float* bv   = sel ? bk : bq;
    const int coff    = sel ? DD : 0;

    float acc = 0.0f;
    for (int e = 0; e < DD; ++e)
        acc = fmaf(W[e * DD + f], Wc[d * (2 * DD) + coff + e], acc);
    (sel ? Ak : Aq)[f * DD + d] = acc * scale;

    if (f == 0) {
        float bacc = 0.0f;
        for (int e = 0; e < DD; ++e)
            bacc = fmaf(bv[e], Wc[d * (2 * DD) + coff + e], bacc);
        bacc *= scale;
        if (!sel) bacc += bc[d];
        (sel ? biask : biasq)[d] = bacc;
    }
}

// ---------------------------------------------------------------------------
// Kernel 1: OUT[2048,128] = X[2048,128] @ A[128,128] + bias, via
// V_WMMA_F32_16X16X4_F32. One 16x16 C tile per wave, K-loop of 32 steps.
// A-frag layout (ISA 7.12.2, 32-bit A 16x4): lane<16 holds K={0,1},
// lane>=16 holds K={2,3} -> contiguous float2 load per lane.
// B-frag (4x16, rows striped across lanes per VGPR): vgpr0 = rows K=0/K=2,
// vgpr1 = rows K=1/K=3 for lane halves.
// ---------------------------------------------------------------------------
__global__ __launch_bounds__(256) void proj_wmma_kernel(
    const float* __restrict__ Xq, const float* __restrict__ Xk,
    const float* __restrict__ Aq, const float* __restrict__ Ak,
    const float* __restrict__ biasq, const float* __restrict__ biask,
    float* __restrict__ QC, float* __restrict__ KC) {
    const int sel = blockIdx.y;
    const float* X    = sel ? Xk : Xq;
    const float* A    = sel ? Ak : Aq;
    const float* bias = sel ? biask : biasq;
    float* OUT        = sel ? KC : QC;

    const int wave = threadIdx.x >> 5;
    const int lane = threadIdx.x & 31;
    const int tile = blockIdx.x * 8 + wave;  // 1024 tiles: 128 (M) x 8 (N)
    const int tm   = tile >> 3;
    const int tn   = tile & 7;
    const int hi   = lane >> 4;
    const int m    = lane & 15;
    const int col  = tn * 16 + (lane & 15);

    const float* xrow = X + (size_t)(tm * 16 + m) * DD;

#if __has_builtin(__builtin_amdgcn_wmma_f32_16x16x4_f32)
    v8f c = {};
    for (int k0 = 0; k0 < DD; k0 += 4) {
        const int ka = k0 + 2 * hi;
        v2f a, b;
        a.x = xrow[ka];
        a.y = xrow[ka + 1];
        b.x = A[(size_t)ka * DD + col];
        b.y = A[(size_t)(ka + 1) * DD + col];
        c = __builtin_amdgcn_wmma_f32_16x16x4_f32(
                false, a, false, b, (short)0, c, false, false);
    }
    const float bv = bias[col];
    #pragma unroll
    for (int r = 0; r < 8; ++r) {
        const int row = tm * 16 + r + 8 * hi;
        OUT[(size_t)row * DD + col] = c[r] + bv;
    }
#else
    // Scalar fallback producing identical coverage (should not be taken).
    for (int r = 0; r < 8; ++r) {
        const int row = tm * 16 + r + 8 * hi;
        const float* xr = X + (size_t)row * DD;
        float acc = bias[col];
        for (int k = 0; k < DD; ++k)
            acc = fmaf(xr[k], A[(size_t)k * DD + col], acc);
        OUT[(size_t)row * DD + col] = acc;
    }
#endif
}

// ---------------------------------------------------------------------------
// Kernel 2: one (b,i) row per block. logits[j] = bl + sum_d wl[d]*tanh(qc[d]+kc[j,d]),
// masked softmax with "fully-masked row => unmasked" semantics.
// Trans-pipe bound: ~3 issue slots per element (v_add, v_tanh, v_fma) plus
// two b128 loads per 4 elements.
// ---------------------------------------------------------------------------
__global__ __launch_bounds__(256) void attn_tanh_softmax_kernel(
    const float* __restrict__ QC, const float* __restrict__ KC,
    const int* __restrict__ mask, const float* __restrict__ wl,
    const float* __restrict__ blp, float* __restrict__ out) {
    __shared__ float s_qc[DD];
    __shared__ float s_wl[DD];
    __shared__ float s_red[256];

    const int i = blockIdx.x;
    const int b = blockIdx.y;
    const int t = threadIdx.x;
    const float bl = blp[0];

    if (t < DD) {
        s_qc[t] = QC[((size_t)b * LQ_ + i) * DD + t];
        s_wl[t] = wl[t];
    }
    __syncthreads();

    const float* kcbase = KC + (size_t)b * LKV_ * DD;
    const int*   mrow   = mask + ((size_t)b * LQ_ + i) * LKV_;

    float lg[4];
    int   mv[4];
    int anyValid = 0;
    #pragma unroll
    for (int jj = 0; jj < 4; ++jj) {
        const int j = t + jj * 256;
        const float* kcrow = kcbase + (size_t)j * DD;
        if (jj < 3) __builtin_prefetch(kcbase + (size_t)(j + 256) * DD, 0, 1);
        float acc = bl;
        for (int d = 0; d < DD; d += 4) {
            float4 kc4 = *(const float4*)(kcrow + d);
            float4 qc4 = *(const float4*)(&s_qc[d]);
            float4 wl4 = *(const float4*)(&s_wl[d]);
            float4 xv;
            xv.x = qc4.x + kc4.x;
            xv.y = qc4.y + kc4.y;
            xv.z = qc4.z + kc4.z;
            xv.w = qc4.w + kc4.w;
            float4 tv = dev_tanh4(xv);
            acc = fmaf(wl4.x, tv.x, acc);
            acc = fmaf(wl4.y, tv.y, acc);
            acc = fmaf(wl4.z, tv.z, acc);
            acc = fmaf(wl4.w, tv.w, acc);
        }
        lg[jj] = acc;
        mv[jj] = (mrow[j] != 0);
        anyValid |= mv[jj];
    }
    anyValid = __syncthreads_or(anyValid);

    // mask (unless the whole row is masked), then row max
    float lmax = -__builtin_inff();
    #pragma unroll
    for (int jj = 0; jj < 4; ++jj) {
        if (anyValid && !mv[jj]) lg[jj] = -__builtin_inff();
        lmax = fmaxf(lmax, lg[jj]);
    }
    s_red[t] = lmax;
    __syncthreads();
    for (int s = 128; s > 0; s >>= 1) {
        if (t < s) s_red[t] = fmaxf(s_red[t], s_red[t + s]);
        __syncthreads();
    }
    const float rowmax = s_red[0];
    __syncthreads();

    float ev[4];
    float lsum = 0.0f;
    #pragma unroll
    for (int jj = 0; jj < 4; ++jj) {
        float e = (lg[jj] == -__builtin_inff()) ? 0.0f : dev_exp(lg[jj] - rowmax);
        ev[jj] = e;
        lsum += e;
    }
    s_red[t] = lsum;
    __syncthreads();
    for (int s = 128; s > 0; s >>= 1) {
        if (t < s) s_red[t] += s_red[t + s];
        __syncthreads();
    }
    const float inv = __builtin_amdgcn_rcpf(s_red[0]);

    float* orow = out + ((size_t)b * LQ_ + i) * LKV_;
    #pragma unroll
    for (int jj = 0; jj < 4; ++jj)
        orow[t + jj * 256] = ev[jj] * inv;
}

// ---------------------------------------------------------------------------
extern "C" void kernel_launch(void* const* d_in, const int* in_sizes, int n_in,
                              void* d_out, int out_size, void* d_ws, size_t ws_size,
                              hipStream_t stream) {
    const float* query = (const float*)d_in[0];
    const float* key   = (const float*)d_in[1];
    /* d_in[2] value: projected v is never used in the returned weights */
    const int*   mask  = (const int*)d_in[3];
    const float* Wq    = (const float*)d_in[4];
    const float* bq    = (const float*)d_in[5];
    const float* Wk    = (const float*)d_in[6];
    const float* bk    = (const float*)d_in[7];
    /* d_in[8] Wv, d_in[9] bv: unused */
    const float* Wc    = (const float*)d_in[10];
    const float* bc    = (const float*)d_in[11];
    const float* wl    = (const float*)d_in[12];
    const float* bl    = (const float*)d_in[13];
    float* out = (float*)d_out;

    // workspace layout (floats): Aq | Ak | biasq | biask | QC | KC  (~2.2 MB)
    float* ws    = (float*)d_ws;
    float* Aq    = ws;
    float* Ak    = Aq + DD * DD;
    float* biasq = Ak + DD * DD;
    float* biask = biasq + DD;
    float* QC    = biask + DD;
    float* KC    = QC + (size_t)BB * LQ_ * DD;

    combine_weights_kernel<<<dim3(DD, 2), DD, 0, stream>>>(
        Wq, bq, Wk, bk, Wc, bc, Aq, Ak, biasq, biask);

    // 2048x128 rows, 16x16 tiles, 8 waves/block -> grid (128, 2)
    proj_wmma_kernel<<<dim3(128, 2), 256, 0, stream>>>(
        query, key, Aq, Ak, biasq, biask, QC, KC);

    attn_tanh_softmax_kernel<<<dim3(LQ_, BB), 256, 0, stream>>>(
        QC, KC, mask, wl, bl, out);
}